// EncoderCrossTransformer_35631048688191
// MI455X (gfx1250) — compile-verified
//
#include <hip/hip_runtime.h>
#include <hip/hip_bf16.h>
#include <math.h>

typedef float v2f __attribute__((ext_vector_type(2)));
typedef float v8f __attribute__((ext_vector_type(8)));

#define B_  16
#define C_  8
#define N_  32
#define L_  64
#define D_  256
#define DK_ 64
#define BSTR 260   // padded LDS row stride (floats): bank-conflict-free fragment loads

// ---------------------------------------------------------------------------
// q[b][k] = sum_d hidden[b][d] * Wq[d][k]
__global__ void q_kernel(const float* __restrict__ hidden,
                         const float* __restrict__ Wq,
                         float* __restrict__ q) {
    int g = blockIdx.x * blockDim.x + threadIdx.x;   // B_*DK_
    if (g >= B_ * DK_) return;
    int b = g / DK_, k = g % DK_;
    float acc = 0.f;
    for (int d = 0; d < D_; ++d)
        acc += hidden[b * D_ + d] * Wq[d * DK_ + k];
    q[g] = acc;
}

// ---------------------------------------------------------------------------
// One workgroup per (b,c,n) node. 512 threads = 16 waves.
// Stages bags (L x D gathered embeddings) and Wc^T in LDS, runs the
// [64x256]x[256x64] GEMM with v_wmma_f32_16x16x4_f32, then tanh/softmax/
// weighted-sum, writing context to out[b,c,n,:].
__global__ __launch_bounds__(512)
void node_attention_kernel(const int*   __restrict__ indices,
                           const int*   __restrict__ cross_lengths,
                           const int*   __restrict__ node_sizes,
                           const float* __restrict__ emb,
                           const float* __restrict__ Wc,
                           const float* __restrict__ bc,
                           const float* __restrict__ v,
                           const float* __restrict__ q,   // [B_, DK_]
                           float*       __restrict__ out) // [B_,C_,N_,D_]
{
    extern __shared__ float smem[];
    float* s_bags = smem;                    // L_ * BSTR
    float* s_wcT  = s_bags + L_ * BSTR;      // DK_ * BSTR  (Wc transposed: [k][d])
    float* s_ep   = s_wcT + DK_ * BSTR;      // L_ * 4 partial energies (per DK-tile)
    float* s_sc   = s_ep + L_ * 4;           // L_ softmax scores

    const int node = blockIdx.x;             // ((b*C + c)*N + n)
    const int n = node % N_;
    const int c = (node / N_) % C_;
    const int b = node / (N_ * C_);
    const int t = threadIdx.x;

    float* outp = out + (size_t)node * D_;

    const bool valid = (c < cross_lengths[b]) && (n < node_sizes[b * C_ + c]);
    if (!valid) {                 // zero bags + zero q => context == 0 exactly
        if (t < D_) outp[t] = 0.0f;
        return;
    }

    // ---- gather embedding rows into LDS (index 0 => padding row == zeros)
    {
        const int l  = t >> 3;               // 0..63
        const int d0 = (t & 7) << 5;         // 0,32,...,224
        const int idx = indices[(size_t)node * L_ + l];
        float4* dst = (float4*)(s_bags + l * BSTR + d0);
        if (idx == 0) {
            float4 z = {0.f, 0.f, 0.f, 0.f};
            #pragma unroll
            for (int j = 0; j < 8; ++j) dst[j] = z;
        } else {
            const float4* src = (const float4*)(emb + (size_t)idx * D_ + d0);
            #pragma unroll
            for (int j = 0; j < 8; ++j) dst[j] = src[j];
        }
    }
    // ---- load Wc transposed into LDS: s_wcT[k][d] = Wc[d][k]
    #pragma unroll 4
    for (int i = 0; i < 32; ++i) {
        int flat = t + i * 512;              // 0..16383, d-major
        int d  = flat >> 6;
        int k  = flat & 63;
        s_wcT[k * BSTR + d] = Wc[flat];
    }
    __syncthreads();

    // ---- WMMA: pre[l][k] tile per wave; wave = (li, ni), li,ni in 0..3
    const int wave = t >> 5;
    const int lane = t & 31;
    const int half = lane >> 4;              // 0 or 1
    const int mr   = lane & 15;
    const int li   = wave >> 2;              // L tile
    const int ni   = wave & 3;               // DK tile

    // A frag: bags[li*16+mr][k0 + 2*half + {0,1}]  (contiguous pair)
    // B frag: Wc[k0+2*half+{0,1}][ni*16+mr] = s_wcT[ni*16+mr ... transposed row]
    const float* aptr = s_bags + (li * 16 + mr) * BSTR + 2 * half;
    const float* bptr = s_wcT  + (ni * 16 + mr) * BSTR + 2 * half;

    v8f acc = {};
    #pragma unroll 8
    for (int k0 = 0; k0 < D_; k0 += 4) {
        v2f a  = *(const v2f*)(aptr + k0);
        v2f bb = *(const v2f*)(bptr + k0);
        acc = __builtin_amdgcn_wmma_f32_16x16x4_f32(
            /*neg_a=*/false, a, /*neg_b=*/false, bb,
            /*c_mod=*/(short)0, acc, /*reuse_a=*/false, /*reuse_b=*/false);
    }

    // acc[i] = pre[l = li*16 + i + 8*half][kk = ni*16 + mr]
    const int kk   = ni * 16 + mr;
    const float bq = bc[kk] + q[b * DK_ + kk];
    const float vk = v[kk];
    #pragma unroll
    for (int i = 0; i < 8; ++i) {
        float e = vk * tanhf(acc[i] + bq);
        // sum over the 16 lanes of this half (kk = ni*16 .. ni*16+15)
        e += __shfl_xor(e, 1, 32);
        e += __shfl_xor(e, 2, 32);
        e += __shfl_xor(e, 4, 32);
        e += __shfl_xor(e, 8, 32);
        if (mr == 0)
            s_ep[(li * 16 + i + 8 * half) * 4 + ni] = e;
    }
    __syncthreads();

    // ---- softmax over L=64 (wave 0; no word mask, as in reference)
    if (wave == 0) {
        int l0 = lane, l1 = lane + 32;
        float e0 = s_ep[l0*4] + s_ep[l0*4+1] + s_ep[l0*4+2] + s_ep[l0*4+3];
        float e1 = s_ep[l1*4] + s_ep[l1*4+1] + s_ep[l1*4+2] + s_ep[l1*4+3];
        float m = fmaxf(e0, e1);
        #pragma unroll
        for (int off = 16; off >= 1; off >>= 1)
            m = fmaxf(m, __shfl_xor(m, off, 32));
        float x0 = expf(e0 - m), x1 = expf(e1 - m);
        float s = x0 + x1;
        #pragma unroll
        for (int off = 16; off >= 1; off >>= 1)
            s += __shfl_xor(s, off, 32);
        float inv = 1.0f / s;
        s_sc[l0] = x0 * inv;
        s_sc[l1] = x1 * inv;
    }
    __syncthreads();

    // ---- context[d] = sum_l score[l] * bags[l][d]
    if (t < D_) {
        float a0 = 0.f;
        #pragma unroll 8
        for (int l = 0; l < L_; ++l)
            a0 += s_sc[l] * s_bags[l * BSTR + t];
        outp[t] = a0;
    }
}

// ---------------------------------------------------------------------------
// cross_score[b][c] = sum over (n,d) of context
__global__ __launch_bounds__(256)
void score_kernel(const float* __restrict__ out, float* __restrict__ score) {
    int bcid = blockIdx.x;                   // B_*C_
    const float* p = out + (size_t)bcid * N_ * D_;
    float acc = 0.f;
    for (int i = threadIdx.x; i < N_ * D_; i += 256) acc += p[i];
    #pragma unroll
    for (int off = 16; off >= 1; off >>= 1) acc += __shfl_xor(acc, off, 32);
    __shared__ float red[8];
    int w = threadIdx.x >> 5, lane = threadIdx.x & 31;
    if (lane == 0) red[w] = acc;
    __syncthreads();
    if (threadIdx.x == 0) {
        float s = 0.f;
        #pragma unroll
        for (int i = 0; i < 8; ++i) s += red[i];
        score[bcid] = s;
    }
}

// ---------------------------------------------------------------------------
// first-argmax over valid groups; zero groups c < best (invalid already zero)
__global__ __launch_bounds__(256)
void mask_kernel(const float* __restrict__ score,
                 const int* __restrict__ cross_lengths,
                 float* __restrict__ out) {
    int b = blockIdx.x;
    __shared__ int s_best;
    if (threadIdx.x == 0) {
        int cl = cross_lengths[b];
        int best = 0;
        float bv = -INFINITY;
        for (int c = 0; c < C_; ++c) {
            float val = (c < cl) ? score[b * C_ + c] : -INFINITY;
            if (val > bv) { bv = val; best = c; }   // strict > => first max
        }
        s_best = best;
    }
    __syncthreads();
    int cnt = s_best * N_ * D_;
    float* p = out + (size_t)b * C_ * N_ * D_;
    for (int i = threadIdx.x; i < cnt; i += 256) p[i] = 0.f;
}

// ---------------------------------------------------------------------------
__global__ void copy_hidden_kernel(const float* __restrict__ hidden,
                                   float* __restrict__ out_tail) {
    int i = blockIdx.x * blockDim.x + threadIdx.x;
    if (i < B_ * D_) out_tail[i] = hidden[i];
}

// ---------------------------------------------------------------------------
extern "C" void kernel_launch(void* const* d_in, const int* in_sizes, int n_in,
                              void* d_out, int out_size, void* d_ws, size_t ws_size,
                              hipStream_t stream) {
    const float* hidden        = (const float*)d_in[0];
    const int*   indices       = (const int*)d_in[1];
    const int*   cross_lengths = (const int*)d_in[2];
    const int*   node_sizes    = (const int*)d_in[3];
    const float* emb           = (const float*)d_in[4];
    const float* Wc            = (const float*)d_in[5];
    const float* bc            = (const float*)d_in[6];
    const float* Wq            = (const float*)d_in[7];
    const float* v             = (const float*)d_in[8];
    float* out = (float*)d_out;

    float* ws_q     = (float*)d_ws;          // B_*DK_
    float* ws_score = ws_q + B_ * DK_;       // B_*C_

    q_kernel<<<(B_ * DK_ + 255) / 256, 256, 0, stream>>>(hidden, Wq, ws_q);

    size_t shmem = (size_t)(L_ * BSTR + DK_ * BSTR + L_ * 4 + L_) * sizeof(float);
    node_attention_kernel<<<B_ * C_ * N_, 512, shmem, stream>>>(
        indices, cross_lengths, node_sizes, emb, Wc, bc, v, ws_q, out);

    score_kernel<<<B_ * C_, 256, 0, stream>>>(out, ws_score);
    mask_kernel<<<B_, 256, 0, stream>>>(ws_score, cross_lengths, out);
    copy_hidden_kernel<<<(B_ * D_ + 255) / 256, 256, 0, stream>>>(
        hidden, out + (size_t)B_ * C_ * N_ * D_);
}